// VerticalLoss_23536420782514
// MI455X (gfx1250) — compile-verified
//
#include <hip/hip_runtime.h>

// CDNA5 / gfx1250, wave32.
// BCE full-reduction: memory-bound streaming kernel.
//   Bytes: 2 x 33.5M x 4B = 268 MB read @ 23.3 TB/s -> ~11.5 us floor.
//   Concurrency: 23.3 TB/s x ~600ns latency ~ 14 MB in flight needed chip-wide;
//   batching 8 b128 loads/wave (~4 KB) before the first wait needs only ~3.5K
//   resident waves to saturate HBM.
// Wave-level reduction via V_WMMA_F32_16X16X4_F32 (fp32 WMMA, exact).

typedef float v2f __attribute__((ext_vector_type(2)));
typedef float v4f __attribute__((ext_vector_type(4)));
typedef float v8f __attribute__((ext_vector_type(8)));

#define BLOCK 256
#define WAVES_PER_BLOCK (BLOCK / 32)
#define VEC_PER_THREAD 8          // 4096 blocks for N = 2^25
#define LOG_CLAMP -100.0f

__device__ __forceinline__ float bce_term(float p, float t) {
    // t*clamp(log p) + (1-t)*clamp(log(1-p))  ==  t*(lp - lq) + lq
    float lp = fmaxf(__logf(p),        LOG_CLAMP);   // p==0 -> -inf -> clamp
    float lq = fmaxf(__logf(1.0f - p), LOG_CLAMP);
    return fmaf(t, lp - lq, lq);
}

__device__ __forceinline__ float bce_vec4(v4f p, v4f t) {
    return (bce_term(p.x, t.x) + bce_term(p.y, t.y))
         + (bce_term(p.z, t.z) + bce_term(p.w, t.w));
}

__global__ void vloss_zero_kernel(float* out) { out[0] = 0.0f; }

__global__ __launch_bounds__(BLOCK) void vloss_bce_kernel(
    const v4f* __restrict__ p4, const v4f* __restrict__ t4,
    const float* __restrict__ ps, const float* __restrict__ ts,
    float* __restrict__ out, unsigned int nvec, unsigned int ntail, float scale)
{
    float acc = 0.0f;
    const unsigned int tid    = blockIdx.x * BLOCK + threadIdx.x;
    const unsigned int stride = gridDim.x * BLOCK;

    // ---- Main loop, 4x unrolled: 8 non-temporal b128 loads batched before use ----
    unsigned int i = tid;
    for (; i + 3u * stride < nvec; i += 4u * stride) {
        v4f p0 = __builtin_nontemporal_load(&p4[i]);
        v4f t0 = __builtin_nontemporal_load(&t4[i]);
        v4f p1 = __builtin_nontemporal_load(&p4[i + stride]);
        v4f t1 = __builtin_nontemporal_load(&t4[i + stride]);
        v4f p2 = __builtin_nontemporal_load(&p4[i + 2u * stride]);
        v4f t2 = __builtin_nontemporal_load(&t4[i + 2u * stride]);
        v4f p3 = __builtin_nontemporal_load(&p4[i + 3u * stride]);
        v4f t3 = __builtin_nontemporal_load(&t4[i + 3u * stride]);
        acc += bce_vec4(p0, t0);
        acc += bce_vec4(p1, t1);
        acc += bce_vec4(p2, t2);
        acc += bce_vec4(p3, t3);
    }
    // Remainder grid-stride iterations (0..3 of them).
    for (; i < nvec; i += stride) {
        v4f p = __builtin_nontemporal_load(&p4[i]);
        v4f t = __builtin_nontemporal_load(&t4[i]);
        acc += bce_vec4(p, t);
    }
    // Scalar tail (N % 4) — empty for the reference shape (N = 2^25).
    if (ntail) {
        unsigned int base = nvec * 4u;
        for (unsigned int k = tid; k < ntail; k += stride)
            acc += bce_term(ps[base + k], ts[base + k]);
    }

    // ---- Wave32 reduction via V_WMMA_F32_16X16X4_F32 ----
    // A (16x4 fp32): a[0] = per-lane partial, a[1] = 0 -> each lane's value in
    // exactly one A slot. B = ones(4x16) -> D[m][*] = row sums. Summing the 8
    // C/D VGPRs gives the half-wave sum (rows 0-7 on lanes 0-15, rows 8-15 on
    // lanes 16-31); one xor-16 permute combines the halves.
    v2f a; a[0] = acc;  a[1] = 0.0f;
    v2f b; b[0] = 1.0f; b[1] = 1.0f;
    v8f c = {};
    c = __builtin_amdgcn_wmma_f32_16x16x4_f32(
        /*neg_a=*/false, a, /*neg_b=*/false, b,
        /*c_mod=*/(short)0, c, /*reuse_a=*/false, /*reuse_b=*/false);
    float half = ((c[0] + c[1]) + (c[2] + c[3])) + ((c[4] + c[5]) + (c[6] + c[7]));
    float wsum = half + __shfl_xor(half, 16, 32);

    // ---- Block reduction: one LDS slot per wave, one atomic per block ----
    __shared__ float smem[WAVES_PER_BLOCK];
    const unsigned int lane = threadIdx.x & 31u;
    const unsigned int wid  = threadIdx.x >> 5;
    if (lane == 0) smem[wid] = wsum;
    __syncthreads();
    if (threadIdx.x == 0) {
        float bsum = 0.0f;
        #pragma unroll
        for (int w = 0; w < WAVES_PER_BLOCK; ++w) bsum += smem[w];
        atomicAdd(out, bsum * scale);
    }
}

extern "C" void kernel_launch(void* const* d_in, const int* in_sizes, int n_in,
                              void* d_out, int out_size, void* d_ws, size_t ws_size,
                              hipStream_t stream) {
    (void)n_in; (void)out_size; (void)d_ws; (void)ws_size;
    const float* p = (const float*)d_in[0];   // vertical_pred, fp32
    const float* t = (const float*)d_in[1];   // vertical_true, fp32
    float* out = (float*)d_out;               // scalar fp32 loss

    const unsigned int N     = (unsigned int)in_sizes[0];  // B*C*H*W = 33,554,432
    const unsigned int nvec  = N / 4u;
    const unsigned int ntail = N % 4u;
    // loss = -(1/(H*B)) * total = -(C/N) * total, C = 8 fixed by the reference.
    const float scale = -8.0f / (float)N;

    vloss_zero_kernel<<<1, 1, 0, stream>>>(out);

    unsigned int blocks = (nvec + BLOCK * VEC_PER_THREAD - 1u) / (BLOCK * VEC_PER_THREAD);
    if (blocks == 0) blocks = 1;               // N = 2^25 -> 4096 blocks
    vloss_bce_kernel<<<blocks, BLOCK, 0, stream>>>(
        (const v4f*)p, (const v4f*)t, p, t, out, nvec, ntail, scale);
}